// Net_60129542660
// MI455X (gfx1250) — compile-verified
//
#include <hip/hip_runtime.h>
#include <math.h>

#define HDIM 256
#define KC   32
#define T_LD 36   // LDS leading dim (floats) for both A and Bt tiles:
                  // fragment reads hit banks ≡0 mod 4 (lanes 0-15) vs ≡2 mod 4
                  // (lanes 16-31) -> conflict-free; 8B/16B alignment preserved.

typedef __attribute__((ext_vector_type(2))) float v2f;
typedef __attribute__((ext_vector_type(8))) float v8f;

// ---------------- degree / norm kernels ----------------

__global__ void deg_init_kernel(float* deg, int n) {
    int i = blockIdx.x * blockDim.x + threadIdx.x;
    if (i < n) deg[i] = 1.0f;   // self-loop contributes 1
}

__global__ void deg_edge_kernel(const int* __restrict__ dst, float* deg, int e) {
    int i = blockIdx.x * blockDim.x + threadIdx.x;
    if (i < e) atomicAdd(&deg[dst[i]], 1.0f);
}

__global__ void deg_rsqrt_kernel(float* deg, int n) {
    int i = blockIdx.x * blockDim.x + threadIdx.x;
    if (i < n) deg[i] = rsqrtf(fmaxf(deg[i], 1.0f));
}

// ---------------- WMMA fp32 GEMM:  C = A[M,K] @ B[K,Nout] (+bias)(+relu) ----------------
// Block: 256 threads = 8 waves; block tile 64(M) x 64(N).
// wm = wave&3 -> 16-row strip; wn = wave>>2 -> 32-col strip; each wave owns
// TWO 16x16 accumulators (A fragment reused by 2 WMMAs).
// B is staged TRANSPOSED in LDS (Bt[col][k]) so both operand fragments are
// contiguous 8-byte pairs -> ds_load_2addr_b64, no repack movs.
// OOB rows/cols are CLAMPED during staging (garbage only feeds unstored C
// elements) -> branch-free staging. K is a multiple of KC (128/256): no K guards.

__global__ __launch_bounds__(256)
void gemm_wmma_kernel(const float* __restrict__ A, const float* __restrict__ B,
                      const float* __restrict__ bias, float* __restrict__ C,
                      int M, int K, int Nout, int act)
{
    __shared__ float As[64 * T_LD];   // A tile:  [row][k]   9216 B
    __shared__ float Bt[64 * T_LD];   // B tile:  [col][k]   9216 B (transposed)

    const int tid   = threadIdx.x;
    const int lane  = tid & 31;
    const int wave  = tid >> 5;
    const int wm    = wave & 3;        // 0..3  (M sub-tile)
    const int wn    = wave >> 2;       // 0..1  (N sub-strip of 32)
    const int rowBase = blockIdx.x * 64;
    const int colBase = blockIdx.y * 64;

    // ISA 7.12.2 fragment layouts (16x16x4 f32, wave32):
    //  A: lane L -> row M = L%16, VGPR0/1 = K = 2*(L/16), 2*(L/16)+1
    //  B: lane L -> col N = L%16, VGPR0/1 = K = 2*(L/16), 2*(L/16)+1
    //  C/D: lane L -> col N = L%16, VGPR v -> row M = v + 8*(L/16)
    const int frow  = lane & 15;
    const int khalf = lane >> 4;

    // staging decomposition
    const int ar   = tid >> 3;          // A row 0..31 (then +32)
    const int ac4  = (tid & 7) * 4;     // A k-offset {0,4,...,28}
    const int bcol = tid & 63;          // B column within tile
    const int bkq  = tid >> 6;          // B k-quad 0..3 (then +4)

    const int Mm1 = M - 1;
    const int Nm1 = Nout - 1;

    v8f acc0, acc1;
#pragma unroll
    for (int i = 0; i < 8; ++i) { acc0[i] = 0.0f; acc1[i] = 0.0f; }

    int gc = colBase + bcol; if (gc > Nm1) gc = Nm1;   // column clamp (once)

    for (int k0 = 0; k0 < K; k0 += KC) {
        // speculative prefetch of next A K-chunk (global_prefetch_b8)
        if (k0 + KC < K) {
            int pr = rowBase + ar; if (pr > Mm1) pr = Mm1;
            __builtin_prefetch(A + (size_t)pr * K + (k0 + KC) + ac4, 0, 1);
        }

        // stage A tile 64 x KC  (row-clamped, b128 load + b128 LDS store)
#pragma unroll
        for (int it = 0; it < 2; ++it) {
            int r  = ar + it * 32;
            int gr = rowBase + r; if (gr > Mm1) gr = Mm1;
            float4 v = *(const float4*)(A + (size_t)gr * K + k0 + ac4);
            *(float4*)(&As[r * T_LD + ac4]) = v;
        }
        // stage B tile KC x 64 transposed: each thread walks one column,
        // 4 coalesced b32 row loads -> one b128 LDS store
#pragma unroll
        for (int it = 0; it < 2; ++it) {
            int kq = bkq + it * 4;              // 0..7
            const float* Bp = B + (size_t)(k0 + kq * 4) * Nout + gc;
            float4 v;
            v.x = Bp[0];
            v.y = Bp[(size_t)Nout];
            v.z = Bp[(size_t)Nout * 2];
            v.w = Bp[(size_t)Nout * 3];
            *(float4*)(&Bt[bcol * T_LD + kq * 4]) = v;
        }
        __syncthreads();

#pragma unroll
        for (int kk = 0; kk < KC; kk += 4) {
            const int kb = kk + 2 * khalf;
            v2f a  = *(const v2f*)(&As[(16 * wm + frow) * T_LD + kb]);
            const float* bp = &Bt[(32 * wn + frow) * T_LD + kb];
            v2f b0 = *(const v2f*)(bp);
            v2f b1 = *(const v2f*)(bp + 16 * T_LD);
            // 8 args: (neg_a, A, neg_b, B, c_mod, C, reuse_a, reuse_b)
            acc0 = __builtin_amdgcn_wmma_f32_16x16x4_f32(
                false, a, false, b0, (short)0, acc0, false, false);
            acc1 = __builtin_amdgcn_wmma_f32_16x16x4_f32(
                false, a, false, b1, (short)0, acc1, false, false);
        }
        __syncthreads();
    }

    const int col0 = colBase + 32 * wn + frow;
    const int col1 = col0 + 16;
    float bv0 = 0.0f, bv1 = 0.0f;
    if (bias) {
        if (col0 < Nout) bv0 = bias[col0];
        if (col1 < Nout) bv1 = bias[col1];
    }
#pragma unroll
    for (int v = 0; v < 8; ++v) {
        int row = rowBase + 16 * wm + 8 * khalf + v;
        if (row < M) {
            if (col0 < Nout) {
                float r = acc0[v] + bv0;
                if (act) r = fmaxf(r, 0.0f);
                C[(size_t)row * Nout + col0] = r;
            }
            if (col1 < Nout) {
                float r = acc1[v] + bv1;
                if (act) r = fmaxf(r, 0.0f);
                C[(size_t)row * Nout + col1] = r;
            }
        }
    }
}

// ---------------- aggregation kernels ----------------

// agg[n,:] = h[n,:] * dinv[n]^2   (self-loop message)
__global__ void self_msg_kernel(const float* __restrict__ h, const float* __restrict__ dinv,
                                float* __restrict__ agg, int n)
{
    size_t idx = (size_t)blockIdx.x * blockDim.x + threadIdx.x;
    size_t total = (size_t)n * (HDIM / 4);
    if (idx >= total) return;
    int node = (int)(idx >> 6);             // HDIM/4 == 64
    int f4   = (int)(idx & 63) * 4;
    float w = dinv[node]; w *= w;
    const float4 hv = *(const float4*)(h + (size_t)node * HDIM + f4);
    float4 o; o.x = hv.x * w; o.y = hv.y * w; o.z = hv.z * w; o.w = hv.w * w;
    *(float4*)(agg + (size_t)node * HDIM + f4) = o;
}

// agg[dst,:] += h[src,:] * dinv[src]*dinv[dst]   (64 threads/edge, 4 feats/thread)
__global__ void edge_scatter_kernel(const float* __restrict__ h,
                                    const int* __restrict__ src, const int* __restrict__ dst,
                                    const float* __restrict__ dinv,
                                    float* __restrict__ agg, int e)
{
    size_t idx = (size_t)blockIdx.x * blockDim.x + threadIdx.x;
    size_t total = (size_t)e * 64;
    if (idx >= total) return;
    int ed = (int)(idx >> 6);
    int f4 = (int)(idx & 63) * 4;
    int s = src[ed], d = dst[ed];
    float w = dinv[s] * dinv[d];
    const float4 hv = *(const float4*)(h + (size_t)s * HDIM + f4);
    float* ap = agg + (size_t)d * HDIM + f4;
    atomicAdd(ap + 0, hv.x * w);
    atomicAdd(ap + 1, hv.y * w);
    atomicAdd(ap + 2, hv.z * w);
    atomicAdd(ap + 3, hv.w * w);
}

// y = relu(agg + b)
__global__ void bias_relu_kernel(const float* __restrict__ agg, const float* __restrict__ b,
                                 float* __restrict__ y, int n)
{
    size_t idx = (size_t)blockIdx.x * blockDim.x + threadIdx.x;
    size_t total = (size_t)n * (HDIM / 4);
    if (idx >= total) return;
    int node = (int)(idx >> 6);
    int f4   = (int)(idx & 63) * 4;
    const float4 av = *(const float4*)(agg + (size_t)node * HDIM + f4);
    float4 o;
    o.x = fmaxf(av.x + b[f4 + 0], 0.0f);
    o.y = fmaxf(av.y + b[f4 + 1], 0.0f);
    o.z = fmaxf(av.z + b[f4 + 2], 0.0f);
    o.w = fmaxf(av.w + b[f4 + 3], 0.0f);
    *(float4*)(y + (size_t)node * HDIM + f4) = o;
}

// ---------------- launch ----------------

extern "C" void kernel_launch(void* const* d_in, const int* in_sizes, int n_in,
                              void* d_out, int out_size, void* d_ws, size_t ws_size,
                              hipStream_t stream)
{
    (void)n_in; (void)ws_size;

    const float* x     = (const float*)d_in[0];
    const int*   ei    = (const int*)d_in[1];
    const float* W0    = (const float*)d_in[2];  const float* b0    = (const float*)d_in[3];
    const float* W1    = (const float*)d_in[4];  const float* b1    = (const float*)d_in[5];
    const float* W2    = (const float*)d_in[6];  const float* b2    = (const float*)d_in[7];
    const float* W3    = (const float*)d_in[8];  const float* b3    = (const float*)d_in[9];
    const float* fc1_w = (const float*)d_in[10]; const float* fc1_b = (const float*)d_in[11];
    const float* fc2_w = (const float*)d_in[12]; const float* fc2_b = (const float*)d_in[13];
    const float* fc3_w = (const float*)d_in[14]; const float* fc3_b = (const float*)d_in[15];

    const int N = in_sizes[0] / 128;     // 50000
    const int E = in_sizes[1] / 2;       // 800000
    const int F_IN = 128;
    const int Cdim = out_size / N;       // 121

    const int* src = ei;
    const int* dst = ei + E;

    // workspace: dinv[Na] | h[N*H] | agg[N*H] | y[N*H]
    float* ws   = (float*)d_ws;
    size_t Na   = ((size_t)N + 63) & ~(size_t)63;
    float* dinv = ws;
    float* h    = ws + Na;
    float* agg  = h + (size_t)N * HDIM;
    float* ybuf = agg + (size_t)N * HDIM;

    const dim3 blk(256);
    const int nBlocks  = (N + 255) / 256;
    const int eBlocks  = (E + 255) / 256;
    const size_t nh4   = (size_t)N * (HDIM / 4);
    const int nhBlocks = (int)((nh4 + 255) / 256);
    const size_t es    = (size_t)E * 64;
    const int esBlocks = (int)((es + 255) / 256);

    // symmetric normalization
    deg_init_kernel <<<nBlocks, blk, 0, stream>>>(dinv, N);
    deg_edge_kernel <<<eBlocks, blk, 0, stream>>>(dst, dinv, E);
    deg_rsqrt_kernel<<<nBlocks, blk, 0, stream>>>(dinv, N);

    const dim3 gH((N + 63) / 64, (HDIM + 63) / 64);

    const float* Wl[4] = {W0, W1, W2, W3};
    const float* bl[4] = {b0, b1, b2, b3};
    const float* in = x;
    int Kin = F_IN;
    for (int l = 0; l < 4; ++l) {
        gemm_wmma_kernel   <<<gH, blk, 0, stream>>>(in, Wl[l], nullptr, h, N, Kin, HDIM, 0);
        self_msg_kernel    <<<nhBlocks, blk, 0, stream>>>(h, dinv, agg, N);
        edge_scatter_kernel<<<esBlocks, blk, 0, stream>>>(h, src, dst, dinv, agg, E);
        bias_relu_kernel   <<<nhBlocks, blk, 0, stream>>>(agg, bl[l], ybuf, N);
        in = ybuf; Kin = HDIM;
    }

    // MLP head
    gemm_wmma_kernel<<<gH, blk, 0, stream>>>(ybuf, fc1_w, fc1_b, h,    N, HDIM, HDIM, 1);
    gemm_wmma_kernel<<<gH, blk, 0, stream>>>(h,    fc2_w, fc2_b, ybuf, N, HDIM, HDIM, 1);

    const dim3 gC((N + 63) / 64, (Cdim + 63) / 64);
    gemm_wmma_kernel<<<gC, blk, 0, stream>>>(ybuf, fc3_w, fc3_b, (float*)d_out, N, HDIM, Cdim, 0);
}